// TimeAttention_68582037782609
// MI455X (gfx1250) — compile-verified
//
#include <hip/hip_runtime.h>

typedef __attribute__((ext_vector_type(2))) float v2f;
typedef __attribute__((ext_vector_type(8))) float v8f;

#define NH   16
#define HD   64
#define EDIM 1024
#define E3   3072
#define TPW  16   // tokens per workgroup

// Fused: qkv GEMM -> RoPE -> 16x16 head-attention -> proj GEMM, all fp32 WMMA.
// X tile is brought into LDS with CDNA5 async global->LDS copies (ASYNCcnt path).
__global__ __launch_bounds__(256, 1)
void time_attn_fused(const float* __restrict__ x,
                     const float* __restrict__ qkv_w,
                     const float* __restrict__ qkv_b,
                     const float* __restrict__ proj_w,
                     const float* __restrict__ proj_b,
                     float* __restrict__ out)
{
    __shared__ float sX[TPW * EDIM];    // 64 KB; reused as attention-output tile
    __shared__ float sQKV[TPW * E3];    // 192 KB
    __shared__ float sS[8 * 256];       // 8 KB: per-wave 16x16 score tile

    const int tid  = threadIdx.x;
    const int lane = tid & 31;
    const int wid  = tid >> 5;          // 8 waves
    const int n16  = lane & 15;
    const int g    = lane >> 4;         // lane half (K-pair select per ISA layout)
    const long tok0 = (long)blockIdx.x * TPW;

    // ---------------- async load X tile (16 x 1024 fp32 = 64 KB) into LDS ----------------
    // global_load_async_to_lds_b128: per-lane 16B global -> LDS, tracked by ASYNCcnt,
    // no VGPR staging. GVS mode: SGPR base + 32-bit VGPR byte offset.
    {
        const float* xbase = x + tok0 * EDIM;                 // uniform -> SGPR pair
        const unsigned lds_base = (unsigned)(uintptr_t)sX;    // low 32 bits = LDS offset
        #pragma unroll 4
        for (int i = tid; i < TPW * EDIM / 4; i += 256) {     // 4096 float4 chunks
            unsigned goff = (unsigned)i * 16u;                // byte offset in tile
            unsigned loff = lds_base + (unsigned)i * 16u;     // LDS byte address
            asm volatile("global_load_async_to_lds_b128 %0, %1, %2"
                         :: "v"(loff), "v"(goff), "s"(xbase)
                         : "memory");
        }
        asm volatile("s_wait_asynccnt 0" ::: "memory");
    }
    __syncthreads();

    // ---------------- GEMM1: qkv = X @ Wqkv^T + b  (M=16, N=3072, K=1024) ----------------
    for (int nt = wid; nt < E3 / 16; nt += 8) {
        const int n0 = nt * 16;
        const float* __restrict__ Wrow = qkv_w + (size_t)(n0 + n16) * EDIM;
        // warm L2 for the next N-tile this wave will process (global_prefetch_b8)
        if (nt + 8 < E3 / 16)
            __builtin_prefetch(qkv_w + (size_t)(n0 + 128 + n16) * EDIM, 0, 0);
        const float* Arow = sX + n16 * EDIM;
        v8f c = {};
        #pragma unroll 4
        for (int k0 = 0; k0 < EDIM; k0 += 4) {
            const int kk = k0 + 2 * g;
            v2f a = { Arow[kk], Arow[kk + 1] };
            v2f b = { Wrow[kk], Wrow[kk + 1] };
            c = __builtin_amdgcn_wmma_f32_16x16x4_f32(false, a, false, b,
                                                      (short)0, c, false, false);
        }
        const float bias = qkv_b[n0 + n16];
        #pragma unroll
        for (int r = 0; r < 8; ++r)
            sQKV[(r + 8 * g) * E3 + n0 + n16] = c[r] + bias;   // C row m = r + 8*(lane/16)
    }
    __syncthreads();

    // ---------------- RoPE on q and k (positions = head index 0..15) ----------------
    // out[2p]   = x[2p]*cos - x[2p+1]*sin ; out[2p+1] = x[2p+1]*cos + x[2p]*sin
    for (int idx = tid; idx < 2 * TPW * NH * 32; idx += 256) {
        const int p  = idx & 31;          // pair index   (d/2 = 32)
        const int i  = (idx >> 5) & 15;   // head = rope position
        const int m  = (idx >> 9) & 15;   // token row
        const int qk = idx >> 13;         // 0 = q, 1 = k
        const float invf = __expf(-(float)p * (9.210340371976184f / 32.f)); // 10000^(-p/32)
        float sn, cs;
        __sincosf((float)i * invf, &sn, &cs);
        float* bp = &sQKV[m * E3 + qk * EDIM + i * HD + 2 * p];
        const float x0 = bp[0], x1 = bp[1];
        bp[0] = x0 * cs - x1 * sn;
        bp[1] = x1 * cs + x0 * sn;
    }
    __syncthreads();

    // ---------------- attention: one token per wave, 2 rounds (uniform barriers) -------
    for (int m = wid; m < TPW; m += 8) {
        const float* q = &sQKV[m * E3];
        const float* k = &sQKV[m * E3 + EDIM];
        const float* v = &sQKV[m * E3 + 2 * EDIM];
        float* S = &sS[wid * 256];

        // S = q @ k^T  (16x16, K=64)
        v8f s = {};
        #pragma unroll
        for (int k0 = 0; k0 < HD; k0 += 4) {
            const int kk = k0 + 2 * g;
            v2f a = { q[n16 * HD + kk], q[n16 * HD + kk + 1] };
            v2f b = { k[n16 * HD + kk], k[n16 * HD + kk + 1] };
            s = __builtin_amdgcn_wmma_f32_16x16x4_f32(false, a, false, b,
                                                      (short)0, s, false, false);
        }
        #pragma unroll
        for (int r = 0; r < 8; ++r)
            S[(r + 8 * g) * 16 + n16] = s[r] * 0.125f;   // * 1/sqrt(64)
        __syncthreads();

        // softmax over each row (lanes 0..15, one row each)
        if (lane < 16) {
            float* row = &S[lane * 16];
            float mx = row[0];
            #pragma unroll
            for (int j = 1; j < 16; ++j) mx = fmaxf(mx, row[j]);
            float sum = 0.f;
            #pragma unroll
            for (int j = 0; j < 16; ++j) { float e = __expf(row[j] - mx); row[j] = e; sum += e; }
            const float inv = 1.f / sum;
            #pragma unroll
            for (int j = 0; j < 16; ++j) row[j] *= inv;
        }
        __syncthreads();

        // O = S @ v  (16x64, K=16), written into sX as the proj-GEMM A tile
        #pragma unroll
        for (int d0 = 0; d0 < HD; d0 += 16) {
            v8f o = {};
            #pragma unroll
            for (int k0 = 0; k0 < 16; k0 += 4) {
                const int kk = k0 + 2 * g;
                v2f a = { S[n16 * 16 + kk], S[n16 * 16 + kk + 1] };
                v2f b = { v[kk * HD + d0 + n16], v[(kk + 1) * HD + d0 + n16] };
                o = __builtin_amdgcn_wmma_f32_16x16x4_f32(false, a, false, b,
                                                          (short)0, o, false, false);
            }
            #pragma unroll
            for (int r = 0; r < 8; ++r)
                sX[m * EDIM + (r + 8 * g) * HD + d0 + n16] = o[r];
        }
    }
    __syncthreads();

    // ---------------- proj GEMM: out = O @ Pw^T + pb  (M=16, N=1024, K=1024) ----------
    for (int nt = wid; nt < EDIM / 16; nt += 8) {
        const int n0 = nt * 16;
        const float* __restrict__ Wrow = proj_w + (size_t)(n0 + n16) * EDIM;
        if (nt + 8 < EDIM / 16)
            __builtin_prefetch(proj_w + (size_t)(n0 + 128 + n16) * EDIM, 0, 0);
        const float* Arow = sX + n16 * EDIM;
        v8f c = {};
        #pragma unroll 4
        for (int k0 = 0; k0 < EDIM; k0 += 4) {
            const int kk = k0 + 2 * g;
            v2f a = { Arow[kk], Arow[kk + 1] };
            v2f b = { Wrow[kk], Wrow[kk + 1] };
            c = __builtin_amdgcn_wmma_f32_16x16x4_f32(false, a, false, b,
                                                      (short)0, c, false, false);
        }
        const float bias = proj_b[n0 + n16];
        #pragma unroll
        for (int r = 0; r < 8; ++r)
            out[(tok0 + r + 8 * g) * EDIM + n0 + n16] = c[r] + bias;
    }
}

extern "C" void kernel_launch(void* const* d_in, const int* in_sizes, int n_in,
                              void* d_out, int out_size, void* d_ws, size_t ws_size,
                              hipStream_t stream) {
    (void)in_sizes; (void)n_in; (void)out_size; (void)d_ws; (void)ws_size;
    const float* x      = (const float*)d_in[0];
    const float* qkv_w  = (const float*)d_in[1];
    const float* qkv_b  = (const float*)d_in[2];
    const float* proj_w = (const float*)d_in[3];
    const float* proj_b = (const float*)d_in[4];
    float* out = (float*)d_out;

    const int n_tokens = 2 * 16 * 32 * 32;          // b*t*h*w = 32768
    dim3 grid(n_tokens / TPW), block(256);
    time_attn_fused<<<grid, block, 0, stream>>>(x, qkv_w, qkv_b, proj_w, proj_b, out);
}